// AttentionFusion_17712445129136
// MI455X (gfx1250) — compile-verified
//
#include <hip/hip_runtime.h>
#include <hip/hip_bf16.h>
#include <math.h>

typedef __attribute__((ext_vector_type(2))) float v2f;
typedef __attribute__((ext_vector_type(8))) float v8f;
typedef int v4i_vs __attribute__((vector_size(16)));   // matches builtin pointee type

#define D_DIM 128
#define K_STEPS 32                 // 128 / 4
#define TILE_M 16                  // rain rows per staged tile
#define TILE_STRIDE 132            // padded row stride (dwords) -> conflict-free banks
#define HY2_OFF (TILE_M * TILE_STRIDE)
#define BUF_FLOATS (HY2_OFF + TILE_M)
#define COLS_PER_BLOCK 256         // 8 waves * 32 clear rows
#define MSPLIT 8                   // M-dimension block split

#define GLOBAL_AS __attribute__((address_space(1)))
#define LDS_AS    __attribute__((address_space(3)))

#if __has_builtin(__builtin_amdgcn_global_load_async_to_lds_b128) && \
    __has_builtin(__builtin_amdgcn_global_load_async_to_lds_b32)
#define HAVE_ASYNC_LDS 1
#endif

__device__ __forceinline__ void copy_b128_to_lds(float* lds, const float* g) {
#ifdef HAVE_ASYNC_LDS
    __builtin_amdgcn_global_load_async_to_lds_b128(
        (GLOBAL_AS v4i_vs*)(const_cast<float*>(g)),
        (LDS_AS v4i_vs*)lds, 0, 0);
#else
    *reinterpret_cast<float4*>(lds) = *reinterpret_cast<const float4*>(g);
#endif
}

__device__ __forceinline__ void copy_b32_to_lds(float* lds, const float* g) {
#ifdef HAVE_ASYNC_LDS
    __builtin_amdgcn_global_load_async_to_lds_b32(
        (GLOBAL_AS int*)(const_cast<float*>(g)),
        (LDS_AS int*)lds, 0, 0);
#else
    *lds = *g;
#endif
}

__device__ __forceinline__ void wait_async_lds() {
#ifdef HAVE_ASYNC_LDS
#if __has_builtin(__builtin_amdgcn_s_wait_asynccnt)
    __builtin_amdgcn_s_wait_asynccnt(0);
#else
    asm volatile("s_wait_asynccnt 0x0" ::: "memory");
#endif
#endif
}

// Monotonic (score, index) packing: bigger score wins; on ties smaller index wins.
__device__ __forceinline__ unsigned long long pack_score(float sc, int idx) {
    unsigned u = __float_as_uint(sc);
    u = (u & 0x80000000u) ? ~u : (u | 0x80000000u);
    return ((unsigned long long)u << 32) | (unsigned)(0xFFFFFFFFu - (unsigned)idx);
}

// ---------------------------------------------------------------------------
// Kernel 0: hy2[m] = 0.5*||rain[m]||^2  and  best[n] = 0 (atomic buffer init).
// ---------------------------------------------------------------------------
__global__ void prep_kernel(const float* __restrict__ rain,
                            float* __restrict__ hy2,
                            unsigned long long* __restrict__ best,
                            int M, int N) {
    int i = blockIdx.x * blockDim.x + threadIdx.x;
    if (i < M) {
        const float4* rp = reinterpret_cast<const float4*>(rain + (size_t)i * D_DIM);
        float s = 0.f;
#pragma unroll
        for (int t = 0; t < D_DIM / 4; ++t) {
            float4 v = rp[t];
            s += v.x * v.x + v.y * v.y + v.z * v.z + v.w * v.w;
        }
        hy2[i] = 0.5f * s;
    }
    if (i < N) best[i] = 0ull;
}

// ---------------------------------------------------------------------------
// Kernel 1: fused score-GEMM + argmax.
//   score[i][j] = clear_i . rain_j - 0.5*||rain_j||^2
// D = A(rain 16xK) x B(clear^T Kx16): lane = clear row (N), VGPR = rain row (M).
// Each wave keeps TWO clear col-tiles (B in regs) -> 2 independent WMMA chains
// sharing each A fragment. Rain tiles double-buffered in LDS via async loads.
// ---------------------------------------------------------------------------
__global__ void __launch_bounds__(256)
knn_kernel(const float* __restrict__ clear_f,
           const float* __restrict__ rain_f,
           const float* __restrict__ hy2,
           unsigned long long* __restrict__ best,
           int N, int M) {
    __shared__ float buf[2][BUF_FLOATS];

    const int tid  = threadIdx.x;
    const int wave = tid >> 5;
    const int lane = tid & 31;
    const int col  = lane & 15;          // clear column within a 16-tile
    const int hi   = (lane >> 4) & 1;    // K-half selector for fragments

    const int n0 = blockIdx.x * COLS_PER_BLOCK + wave * 32;
    const int na = n0 + col;
    const int nb = n0 + 16 + col;
    const int nca = na < N ? na : (N - 1);
    const int ncb = nb < N ? nb : (N - 1);

    // Preload clear-side B fragments (2 tiles x 32 x float2 = 128 VGPRs).
    const float* ca = clear_f + (size_t)nca * D_DIM;
    const float* cb = clear_f + (size_t)ncb * D_DIM;
    v2f B0[K_STEPS], B1[K_STEPS];
#pragma unroll
    for (int s = 0; s < K_STEPS; ++s) {
        int k = 4 * s + 2 * hi;
        B0[s] = *reinterpret_cast<const v2f*>(ca + k);
        B1[s] = *reinterpret_cast<const v2f*>(cb + k);
    }

    // This block's chunk of rain tiles (M split across blockIdx.y).
    const int tiles_total = (M + TILE_M - 1) / TILE_M;
    const int tiles_per   = (tiles_total + MSPLIT - 1) / MSPLIT;
    const int t_begin = blockIdx.y * tiles_per;
    const int t_end   = min(t_begin + tiles_per, tiles_total);
    if (t_begin >= t_end) return;   // uniform per block

    // Stage one 16x128 rain tile (+16 hy2 values) into buffer b.
    auto stage = [&](int tile, int b) {
        const int m0 = tile * TILE_M;
#pragma unroll
        for (int q = 0; q < 2; ++q) {
            int seg  = tid * 2 + q;          // 512 x 16B segments
            int row  = seg >> 5;
            int cpos = seg & 31;
            int mrow = m0 + row;
            if (mrow >= M) mrow = M - 1;
            copy_b128_to_lds(&buf[b][row * TILE_STRIDE + cpos * 4],
                             rain_f + (size_t)mrow * D_DIM + cpos * 4);
        }
        if (tid < TILE_M) {
            int mm = m0 + tid;
            if (mm >= M) mm = M - 1;
            copy_b32_to_lds(&buf[b][HY2_OFF + tid], hy2 + mm);
        }
    };

    stage(t_begin, 0);
    wait_async_lds();
    __syncthreads();

    float bv0 = -INFINITY, bv1 = -INFINITY;
    int   bi0 = 0,          bi1 = 0;
    int cur = 0;

    for (int t = t_begin; t < t_end; ++t, cur ^= 1) {
        if (t + 1 < t_end) stage(t + 1, cur ^ 1);   // overlap with compute

        v8f acc0 = {}, acc1 = {};
#pragma unroll
        for (int s = 0; s < K_STEPS; ++s) {
            int k = 4 * s + 2 * hi;
            v2f a = *reinterpret_cast<const v2f*>(&buf[cur][col * TILE_STRIDE + k]);
            acc0 = __builtin_amdgcn_wmma_f32_16x16x4_f32(
                false, a, false, B0[s], (short)0, acc0, false, false);
            acc1 = __builtin_amdgcn_wmma_f32_16x16x4_f32(
                false, a, false, B1[s], (short)0, acc1, false, false);
        }

        const int m0 = t * TILE_M;
#pragma unroll
        for (int r = 0; r < 8; ++r) {
            int m = m0 + r + 8 * hi;
            float hv = buf[cur][HY2_OFF + r + 8 * hi];
            float s0 = acc0[r] - hv;
            float s1 = acc1[r] - hv;
            bool ok = (m < M);
            if (ok && s0 > bv0) { bv0 = s0; bi0 = m; }
            if (ok && s1 > bv1) { bv1 = s1; bi1 = m; }
        }

        wait_async_lds();
        __syncthreads();
    }

    // Combine the two rain-half lanes that own the same clear row.
    {
        float o = __shfl_xor(bv0, 16); int oi = __shfl_xor(bi0, 16);
        if (o > bv0 || (o == bv0 && oi < bi0)) { bv0 = o; bi0 = oi; }
        o = __shfl_xor(bv1, 16); oi = __shfl_xor(bi1, 16);
        if (o > bv1 || (o == bv1 && oi < bi1)) { bv1 = o; bi1 = oi; }
    }
    if (lane < 16) {
        if (na < N) atomicMax(&best[na], pack_score(bv0, bi0));
        if (nb < N) atomicMax(&best[nb], pack_score(bv1, bi1));
    }
}

// ---------------------------------------------------------------------------
// Kernel 2: decode argmax index, gather aligned row, MLP, gated blend.
// One wave per output row.
// ---------------------------------------------------------------------------
__global__ void __launch_bounds__(256)
mlp_fuse_kernel(const float* __restrict__ clear_f,
                const float* __restrict__ rain_f,
                const float* __restrict__ W1,
                const float* __restrict__ b1,
                const float* __restrict__ W2,
                const float* __restrict__ b2,
                const unsigned long long* __restrict__ best,
                float* __restrict__ out, int N) {
    __shared__ float comb[8][2 * D_DIM];

    const int wave = threadIdx.x >> 5;
    const int lane = threadIdx.x & 31;
    const int row  = blockIdx.x * 8 + wave;
    const int r    = row < N ? row : (N - 1);

    const int j = (int)(0xFFFFFFFFu - (unsigned)(best[r] & 0xFFFFFFFFull));
    const float* crow = clear_f + (size_t)r * D_DIM;
    const float* arow = rain_f + (size_t)j * D_DIM;

    float cv[4], av[4];
#pragma unroll
    for (int t = 0; t < 4; ++t) {
        int o = lane + 32 * t;
        cv[t] = crow[o];
        av[t] = arow[o];
        comb[wave][o]         = cv[t];
        comb[wave][D_DIM + o] = av[t];
    }
    __syncthreads();

    float z = 0.f;
    const float4* c4 = reinterpret_cast<const float4*>(comb[wave]);
#pragma unroll
    for (int t = 0; t < 4; ++t) {
        int o = lane + 32 * t;
        float acc = b1[o];
        const float4* wrow = reinterpret_cast<const float4*>(W1 + (size_t)o * 2 * D_DIM);
#pragma unroll 8
        for (int kk = 0; kk < (2 * D_DIM) / 4; ++kk) {
            float4 w = wrow[kk];
            float4 c = c4[kk];
            acc += w.x * c.x + w.y * c.y + w.z * c.z + w.w * c.w;
        }
        float h = fmaxf(acc, 0.f);
        z += h * W2[o];
    }
#pragma unroll
    for (int off = 16; off >= 1; off >>= 1) z += __shfl_xor(z, off);
    z += b2[0];
    float g = 1.f / (1.f + __expf(-z));

    if (row < N) {
#pragma unroll
        for (int t = 0; t < 4; ++t) {
            int o = lane + 32 * t;
            out[(size_t)row * D_DIM + o] = g * cv[t] + (1.f - g) * av[t];
        }
    }
}

// ---------------------------------------------------------------------------
extern "C" void kernel_launch(void* const* d_in, const int* in_sizes, int n_in,
                              void* d_out, int out_size, void* d_ws, size_t ws_size,
                              hipStream_t stream) {
    const float* clear_f = (const float*)d_in[0];
    const float* rain_f  = (const float*)d_in[1];
    const float* W1      = (const float*)d_in[2];
    const float* b1      = (const float*)d_in[3];
    const float* W2      = (const float*)d_in[4];
    const float* b2      = (const float*)d_in[5];
    float* out = (float*)d_out;

    const int N = in_sizes[0] / D_DIM;
    const int M = in_sizes[1] / D_DIM;

    float* hy2 = (float*)d_ws;
    unsigned long long* best =
        (unsigned long long*)((char*)d_ws + (((size_t)M * sizeof(float) + 255) & ~(size_t)255));

    int prep_n = (M > N ? M : N);
    prep_kernel<<<(prep_n + 255) / 256, 256, 0, stream>>>(rain_f, hy2, best, M, N);

    dim3 grid((N + COLS_PER_BLOCK - 1) / COLS_PER_BLOCK, MSPLIT);
    knn_kernel<<<grid, 256, 0, stream>>>(clear_f, rain_f, hy2, best, N, M);

    mlp_fuse_kernel<<<(N + 7) / 8, 256, 0, stream>>>(
        clear_f, rain_f, W1, b1, W2, b2, best, out, N);
}